// PoissonOperators_52536039965341
// MI455X (gfx1250) — compile-verified
//
#include <hip/hip_runtime.h>

typedef __attribute__((ext_vector_type(2))) float v2f;
typedef __attribute__((ext_vector_type(8))) float v8f;

#define PN1 4096
#define PN2 4096

// Each wave: 16 rows x 64 cols (4 x WMMA 16x16x4_f32, A fragment reused).
// Block = 256 threads = 8 waves stacked in M -> 128 rows x 64 cols per block.
// Grid = (PN2/64, PN1/128) = (64, 32).
__global__ __launch_bounds__(256)
void PoissonOperators_52536039965341_kernel(const float* __restrict__ X1,
                                            const float* __restrict__ X2,
                                            const float* __restrict__ ls,
                                            float* __restrict__ out) {
  const int lane = threadIdx.x & 31;
  const int wave = threadIdx.x >> 5;
  const int half = lane >> 4;   // 0: K=0,1 (A) / K-rows 0,1 (B); 1: K=2,3
  const int col  = lane & 15;

  const int m0 = (blockIdx.y * 8 + wave) * 16;  // row-tile base
  const int n0 = blockIdx.x * 64;               // col-strip base

  // gamma = 1 / lengthscale^2 ; result = g2 * exp(-0.5*g*u) * (8 - 8*g*u + g2*u^2)
  const float ell = ls[0];
  const float g   = 1.0f / (ell * ell);
  const float g2  = g * g;

  // ---- A fragment: rows m0..m0+15, augmented row [a0, a1, |a|^2, 1] ----
  // Layout (16x4 f32 A): lanes 0-15: v0=K0, v1=K1 ; lanes 16-31: v0=K2, v1=K3.
  const float ax = X1[(m0 + col) * 2 + 0];
  const float ay = X1[(m0 + col) * 2 + 1];
  v2f afrag;
  afrag.x = half ? (ax * ax + ay * ay) : ax;  // K2 = |a|^2   | K0 = a0
  afrag.y = half ? 1.0f                : ay;  // K3 = 1       | K1 = a1

  for (int t = 0; t < 4; ++t) {
    const int n = n0 + t * 16;

    // ---- B fragment: cols n..n+15, augmented col [-2b0, -2b1, 1, |b|^2] ----
    // Layout (4x16 f32 B): VGPR r, lanes 0-15 -> K=r, lanes 16-31 -> K=2+r.
    const float bx = X2[(n + col) * 2 + 0];
    const float by = X2[(n + col) * 2 + 1];
    v2f bfrag;
    bfrag.x = half ? 1.0f                : (-2.0f * bx);  // K2 = 1    | K0 = -2 b0
    bfrag.y = half ? (bx * bx + by * by) : (-2.0f * by);  // K3=|b|^2  | K1 = -2 b1

    // D = A x B + 0 : D(M,N) = |x1[m0+M] - x2[n+N]|^2
    v8f u = {};
    u = __builtin_amdgcn_wmma_f32_16x16x4_f32(
        /*neg_a=*/false, afrag, /*neg_b=*/false, bfrag,
        /*c_mod=*/(short)0, u, /*reuse_a=*/false, /*reuse_b=*/false);

    // D layout: VGPR r -> row M = 8*half + r, col N = col.
#pragma unroll
    for (int r = 0; r < 8; ++r) {
      const float d2  = u[r];
      const float gu  = g * d2;
      const float val = g2 * __expf(-0.5f * gu)
                           * (8.0f + gu * (-8.0f) + g2 * d2 * d2);
      out[(size_t)(m0 + 8 * half + r) * PN2 + (n + col)] = val;
    }
  }
}

extern "C" void kernel_launch(void* const* d_in, const int* in_sizes, int n_in,
                              void* d_out, int out_size, void* d_ws, size_t ws_size,
                              hipStream_t stream) {
  const float* X1 = (const float*)d_in[0];
  const float* X2 = (const float*)d_in[1];
  const float* ls = (const float*)d_in[2];
  float* out = (float*)d_out;

  dim3 grid(PN2 / 64, PN1 / 128);
  dim3 block(256);
  PoissonOperators_52536039965341_kernel<<<grid, block, 0, stream>>>(X1, X2, ls, out);
}